// Hopf_62311385530394
// MI455X (gfx1250) — compile-verified
//
#include <hip/hip_runtime.h>
#include <cstdint>

// Hopf oscillator bank: B=32, T=2048, U=128.
// Sequential nonlinear scan per (b,u) -> 128 independent wave32 workgroups.
// Inputs streamed via Tensor Data Mover (tensor_load_to_lds, TENSORcnt ring).

#define UNITS   128
#define TSTEPS  2048
#define BATCH   32
#define TILE_U  32                      // units per wave (one wave32 per workgroup)
#define CHUNK   32                      // timesteps per TDM tile
#define NBUF    8                       // ring depth: 8*8KB = 64KB in flight per wave
#define BUF_BYTES    (CHUNK * TILE_U * 4)   // 4 KB per array per buffer
#define CHUNK_GBYTES (CHUNK * UNITS * 4)    // 16 KB global stride per chunk

typedef uint32_t v4u __attribute__((ext_vector_type(4)));
typedef uint32_t v8u __attribute__((ext_vector_type(8)));

// ---- Tensor Data Mover: one instruction DMAs a 32x32-float tile to LDS ----
// D# group 0: [1:0]=count=1, [63:32]=lds_addr, [120:64]=global_addr, [127:126]=type=2
__device__ __forceinline__ void tdm_load(uint32_t lds_addr, uint64_t gaddr, v8u g1) {
    v4u g0;
    g0.x = 1u;                                      // count=1, user mode, no gather
    g0.y = lds_addr;                                // LDS byte address
    g0.z = (uint32_t)gaddr;                         // global_addr[31:0]
    g0.w = (uint32_t)(gaddr >> 32) | 0x80000000u;   // global_addr[56:32] | type=2<<30
    asm volatile("tensor_load_to_lds %0, %1" :: "s"(g0), "s"(g1) : "memory");
}

template <int N>
__device__ __forceinline__ void wait_tensor() {
    asm volatile("s_wait_tensorcnt %0" :: "n"(N) : "memory");
}

__device__ __forceinline__ void wait_ds0() {
    asm volatile("s_wait_dscnt 0" ::: "memory");
}

__global__ __launch_bounds__(TILE_U) void hopf_scan_kernel(
        const float* __restrict__ Xr,
        const float* __restrict__ Xi,
        const float* __restrict__ omegas,
        float* __restrict__ out)
{
    __shared__ float lds_r[NBUF * CHUNK * TILE_U];   // 32 KB
    __shared__ float lds_i[NBUF * CHUNK * TILE_U];   // 32 KB

    const int lane = threadIdx.x;         // 0..31
    const int uq   = blockIdx.x;          // 0..3  : which 32-unit slice
    const int b    = blockIdx.y;          // 0..31 : batch
    const int u    = uq * TILE_U + lane;

    // ---- D# group 1 (constant tile/tensor geometry, all in data_size units) ----
    // data_size=4B; tensor_dim0=128, dim0_stride=128; tensor_dim1=B*T=65536;
    // tile_dim0=32 (units), tile_dim1=32 (timesteps); no pad/iterate/multicast.
    const v8u g1 = {
        (2u << 16),                 // w0: wg_mask=0, data_size=2 (4B)
        (128u << 16),               // w1: tensor_dim0[15:0]=128 @ bits 63:48
        0u,                         // w2: tensor_dim0 hi=0, tensor_dim1 lo=0
        1u | (32u << 16),           // w3: tensor_dim1 hi=1 (65536), tile_dim0=32
        32u,                        // w4: tile_dim1=32, tile_dim2=0
        128u,                       // w5: tensor_dim0_stride=128
        0u,                         // w6: stride0 hi=0, stride1 lo=0
        0u                          // w7: stride1 hi=0
    };

    // Tile origin for this wave: element (b*T + 0, uq*32) of the [B*T, 128] matrix.
    const uint64_t tile_off = ((uint64_t)b * TSTEPS * UNITS + (uint64_t)uq * TILE_U) * 4ull;
    const uint64_t gr_base  = (uint64_t)(uintptr_t)Xr + tile_off;
    const uint64_t gi_base  = (uint64_t)(uintptr_t)Xi + tile_off;

    // Low 32 bits of a generic pointer into LDS == wave-relative LDS byte addr.
    const uint32_t lr_base = (uint32_t)(uintptr_t)(&lds_r[0]);
    const uint32_t li_base = (uint32_t)(uintptr_t)(&lds_i[0]);

    auto issue_chunk = [&](int ch) {
        const uint32_t lofs = (uint32_t)(ch & (NBUF - 1)) * BUF_BYTES;
        const uint64_t gofs = (uint64_t)ch * CHUNK_GBYTES;
        tdm_load(lr_base + lofs, gr_base + gofs, g1);
        tdm_load(li_base + lofs, gi_base + gofs, g1);
    };

    // Prologue: fill the ring (16 TDM ops, 64 KB in flight per wave).
    #pragma unroll
    for (int ch = 0; ch < NBUF; ++ch) issue_chunk(ch);

    // ---- scan state --------------------------------------------------------
    const float omega = omegas[u];
    const float DT    = 1.0f / 173.61f;
    float r   = 1.0f;
    float phi = 0.0f;
    float s   = 0.0f;   // sin(phi)
    float c   = 1.0f;   // cos(phi)

    float* zr = out + (uint64_t)b * TSTEPS * UNITS + u;
    float* zi = zr + (uint64_t)BATCH * TSTEPS * UNITS;

    const int NCHUNK = TSTEPS / CHUNK;    // 64
    for (int ch = 0; ch < NCHUNK; ++ch) {
        // Oldest chunk complete once at most (NBUF-1) chunks (2 ops each) remain.
        wait_tensor<2 * (NBUF - 1)>();    // s_wait_tensorcnt 14

        const int base = (ch & (NBUF - 1)) * (CHUNK * TILE_U);
        #pragma unroll
        for (int t = 0; t < CHUNK; ++t) {
            const float xr = lds_r[base + t * TILE_U + lane];
            const float xi = lds_i[base + t * TILE_U + lane];

            const float input_r   = 0.1f * xr * c;   // uses cos(phi_old)
            const float input_phi = 0.1f * xi * s;   // uses sin(phi_old)

            r   = r   + ((1.0f - 0.01f * r * r) * r + input_r) * DT;
            phi = phi + (omega - input_phi) * DT;

            s = __sinf(phi);                          // v_sin_f32
            c = __cosf(phi);                          // v_cos_f32

            const uint64_t row = (uint64_t)(ch * CHUNK + t) * UNITS;
            __builtin_nontemporal_store(r * c, zr + row);   // z_real
            __builtin_nontemporal_store(r * s, zi + row);   // z_imag
        }

        // All ds reads of this buffer must land before the TDM overwrites it.
        if (ch + NBUF < NCHUNK) {
            wait_ds0();
            issue_chunk(ch + NBUF);
        }
    }
}

extern "C" void kernel_launch(void* const* d_in, const int* in_sizes, int n_in,
                              void* d_out, int out_size, void* d_ws, size_t ws_size,
                              hipStream_t stream) {
    (void)in_sizes; (void)n_in; (void)out_size; (void)d_ws; (void)ws_size;
    const float* Xr     = (const float*)d_in[0];   // [B, T, U]
    const float* Xi     = (const float*)d_in[1];   // [B, T, U]
    const float* omegas = (const float*)d_in[2];   // [1, U]
    float* out          = (float*)d_out;           // z_real ++ z_imag, each [B, T, U]

    dim3 grid(UNITS / TILE_U, BATCH);   // (4, 32) single-wave workgroups
    hopf_scan_kernel<<<grid, TILE_U, 0, stream>>>(Xr, Xi, omegas, out);
}